// MeshLayer_36928128811289
// MI455X (gfx1250) — compile-verified
//
#include <hip/hip_runtime.h>

typedef float v2f __attribute__((ext_vector_type(2)));
typedef float v8f __attribute__((ext_vector_type(8)));

#define N_ 256
#define L_ 256
#define B_ 16384

struct c2 { float r, i; };
__device__ __forceinline__ c2 cexpf2(float t) {
    c2 z;
    __sincosf(t, &z.i, &z.r);
    return z;
}

// ---------------------------------------------------------------------------
// Kernel 0: per-layer complex coefficients diag/off, shape (L, N), stored as
// interleaved float2 in workspace.
//   diag[n] = eps[n]   * 0.25 * (2*psi[n] - psi[n+1] - psi[n-1])
//   off [n] = eps[n-1] * 0.25i * (2*psi[n] + psi[n+1] + psi[n-1])
// with psi[n] = exp(+-i*theta/2) (sign by parity), eps[n] = exp(i*phi) on even
// ports, 1 on odd ports. Indices mod N.
// ---------------------------------------------------------------------------
__global__ __launch_bounds__(256) void mesh_coeffs(
    const float* __restrict__ theta,   // (L, N/2)
    const float* __restrict__ phi,     // (L, N/2)
    float* __restrict__ diag,          // (L*N) float2
    float* __restrict__ off)           // (L*N) float2
{
    int idx = blockIdx.x * 256 + threadIdx.x;   // l*N + n
    int l = idx >> 8;
    int n = idx & 255;
    const float* th = theta + l * (N_ / 2);
    const float* ph = phi   + l * (N_ / 2);

    auto psi = [&](int nn) -> c2 {
        nn &= 255;
        float a = th[nn >> 1] * ((nn & 1) ? -0.5f : 0.5f);
        return cexpf2(a);
    };
    int nm1 = (n + 255) & 255;
    c2 cm = psi(nm1);
    c2 cc = psi(n);
    c2 cp = psi(n + 1);

    c2 en = (n   & 1) ? c2{1.f, 0.f} : cexpf2(ph[n   >> 1]);
    c2 em = (nm1 & 1) ? c2{1.f, 0.f} : cexpf2(ph[nm1 >> 1]);

    float sr = 2.f * cc.r - cp.r - cm.r;
    float si = 2.f * cc.i - cp.i - cm.i;
    diag[2 * idx]     = 0.25f * (en.r * sr - en.i * si);
    diag[2 * idx + 1] = 0.25f * (en.r * si + en.i * sr);

    float ur = 2.f * cc.r + cp.r + cm.r;
    float ui = 2.f * cc.i + cp.i + cm.i;
    off[2 * idx]      = 0.25f * (-em.r * ui - em.i * ur);
    off[2 * idx + 1]  = 0.25f * ( em.r * ur - em.i * ui);
}

// ---------------------------------------------------------------------------
// Kernel 1: fold the entire mesh into one 256x256 complex matrix W by
// propagating each basis vector e_k * exp(i*gamma[k]) through all L layers.
// One workgroup per basis row k, one thread per column j; state lives in LDS.
// Per layer:  t[j] = x[perm[j]];  x'[j] = t[j]*d[j] + t[j^1]*o[j^1].
// The result row W[k][:] is written in WMMA B-fragment order:
//   frag offset = ((ktile*16 + ntile)*32 + lane)*2 + comp
//   lane<16  holds (W[4kt+0][n], W[4kt+1][n]);  lane>=16 holds rows 2,3.
// Also emits the Karatsuba sum fragment Ws = Wr + Wi.
// ---------------------------------------------------------------------------
__global__ __launch_bounds__(256) void build_w(
    const float* __restrict__ gamma,   // (N)
    const int*   __restrict__ perm,    // (L+1, N)
    const float* __restrict__ diag,    // (L*N) float2
    const float* __restrict__ off,     // (L*N) float2
    float* __restrict__ wfr,           // fragment-ordered Re(W)
    float* __restrict__ wfi,           // fragment-ordered Im(W)
    float* __restrict__ wfs)           // fragment-ordered Re(W)+Im(W)
{
    __shared__ float cr[N_];
    __shared__ float ci[N_];
    int k = blockIdx.x;
    int j = threadIdx.x;

    c2 g = cexpf2(gamma[k]);
    cr[j] = (j == k) ? g.r : 0.f;
    ci[j] = (j == k) ? g.i : 0.f;
    __syncthreads();

    int jp = j ^ 1;   // pairwise_perm_idx is [1,0,3,2,...] == j^1
    for (int l = 0; l < L_; ++l) {
        const int* pr = perm + l * N_;
        int p = pr[j];
        int q = pr[jp];
        float ar = cr[p], ai = ci[p];
        float br = cr[q], bi = ci[q];
        int di_ = 2 * (l * N_ + j);
        int oi_ = 2 * (l * N_ + jp);
        float dr = diag[di_], dd = diag[di_ + 1];
        float orr = off[oi_], oo = off[oi_ + 1];
        __syncthreads();
        cr[j] = ar * dr - ai * dd + br * orr - bi * oo;
        ci[j] = ar * dd + ai * dr + br * oo + bi * orr;
        __syncthreads();
    }

    int pf = perm[L_ * N_ + j];        // final right-permutation
    float wr = cr[pf];
    float wi = ci[pf];

    int kt = k >> 2, kin = k & 3, nt = j >> 4;
    int lane = (j & 15) + ((kin >= 2) ? 16 : 0);
    int comp = kin & 1;
    int o = ((kt * 16 + nt) * 32 + lane) * 2 + comp;
    wfr[o] = wr;
    wfi[o] = wi;
    wfs[o] = wr + wi;
}

// ---------------------------------------------------------------------------
// Kernel 2: C = X @ W as a 3-multiplication (Karatsuba) complex f32 WMMA GEMM.
//   P = xr@Wr ; Q = xi@Wi ; T = (xr+xi)@(Wr+Wi)
//   out_re = P - Q ; out_im = T - P - Q
// 8 waves / 256 threads per workgroup; each wave owns a 16-row M slab and
// loops over all 16 N-tiles with a full K=256 reduction (64 x K=4 WMMA steps).
// A fragment (16x4 f32): lane L holds A[L%16][K0 + (L>=16 ? 2:0) + {0,1}].
// B fragments come pre-swizzled from build_w as coalesced float2/lane loads.
// ---------------------------------------------------------------------------
__global__ __launch_bounds__(256) void mesh_gemm(
    const float* __restrict__ xr,      // (B, N)
    const float* __restrict__ xi,      // (B, N)
    const float* __restrict__ wfr,     // fragment-ordered Re(W)
    const float* __restrict__ wfi,     // fragment-ordered Im(W)
    const float* __restrict__ wfs,     // fragment-ordered Re(W)+Im(W)
    float* __restrict__ out)           // (2, B, N)
{
    int tid  = threadIdx.x;
    int wave = tid >> 5;
    int lane = tid & 31;
    int hi   = lane >> 4;              // 0: K 0,1   1: K 2,3
    int lm   = lane & 15;
    int m0   = blockIdx.x * 128 + wave * 16;

    const float* arow = xr + (size_t)(m0 + lm) * N_ + hi * 2;
    const float* brow = xi + (size_t)(m0 + lm) * N_ + hi * 2;
    float* out_re = out;
    float* out_im = out + (size_t)B_ * N_;

    for (int nt = 0; nt < 16; ++nt) {
        v8f P = {}, Q = {}, T = {};
        const float* wur = wfr + nt * 64 + lane * 2;
        const float* wui = wfi + nt * 64 + lane * 2;
        const float* wus = wfs + nt * 64 + lane * 2;
#pragma unroll 8
        for (int k = 0; k < N_; k += 4) {
            v2f a  = *(const v2f*)(arow + k);
            v2f b  = *(const v2f*)(brow + k);
            v2f wu = *(const v2f*)(wur + (k >> 2) * 1024);
            v2f wv = *(const v2f*)(wui + (k >> 2) * 1024);
            v2f ws = *(const v2f*)(wus + (k >> 2) * 1024);
            v2f s  = a + b;
            P = __builtin_amdgcn_wmma_f32_16x16x4_f32(false, a, false, wu,
                                                      (short)0, P, false, false);
            Q = __builtin_amdgcn_wmma_f32_16x16x4_f32(false, b, false, wv,
                                                      (short)0, Q, false, false);
            T = __builtin_amdgcn_wmma_f32_16x16x4_f32(false, s, false, ws,
                                                      (short)0, T, false, false);
        }
        // C/D layout: VGPR v -> (M=v, N=lane) lanes 0-15; (M=v+8, N=lane-16).
        int col   = nt * 16 + lm;
        int rbase = m0 + hi * 8;
#pragma unroll
        for (int vv = 0; vv < 8; ++vv) {
            size_t o  = (size_t)(rbase + vv) * N_ + col;
            float  re = P[vv] - Q[vv];
            out_re[o] = re;
            out_im[o] = T[vv] - P[vv] - Q[vv];
        }
    }
}

// ---------------------------------------------------------------------------
// Host-side launcher. Inputs (setup_inputs order):
//   0: x_re (B,N) f32   1: x_im (B,N) f32   2: theta (L,N/2) f32
//   3: phi (L,N/2) f32  4: gamma (N) f32    5: perm_idx (L+1,N) i32
//   6: pairwise_perm_idx (N) i32 (== j^1, hardcoded)
// Workspace (floats): diag[131072] | off[131072] | wfr[65536] | wfi[65536] |
//                     wfs[65536]   -> 458752 floats = 1.75 MiB
// ---------------------------------------------------------------------------
extern "C" void kernel_launch(void* const* d_in, const int* in_sizes, int n_in,
                              void* d_out, int out_size, void* d_ws, size_t ws_size,
                              hipStream_t stream)
{
    (void)in_sizes; (void)n_in; (void)out_size; (void)ws_size;
    const float* x_re  = (const float*)d_in[0];
    const float* x_im  = (const float*)d_in[1];
    const float* theta = (const float*)d_in[2];
    const float* phi   = (const float*)d_in[3];
    const float* gamma = (const float*)d_in[4];
    const int*   perm  = (const int*)  d_in[5];

    float* ws   = (float*)d_ws;
    float* diag = ws;                       // 2*L*N floats
    float* off  = ws + 2 * L_ * N_;         // 2*L*N floats
    float* wfr  = ws + 4 * L_ * N_;         // L*N floats (fragment order)
    float* wfi  = wfr + L_ * N_;            // L*N floats
    float* wfs  = wfi + L_ * N_;            // L*N floats

    mesh_coeffs<<<(L_ * N_) / 256, 256, 0, stream>>>(theta, phi, diag, off);
    build_w<<<N_, N_, 0, stream>>>(gamma, perm, diag, off, wfr, wfi, wfs);
    mesh_gemm<<<B_ / 128, 256, 0, stream>>>(x_re, x_im, wfr, wfi, wfs,
                                            (float*)d_out);
}